// MeanAggregator_28424093564967
// MI455X (gfx1250) — compile-verified
//
#include <hip/hip_runtime.h>
#include <stdint.h>

#define D_FEAT 128
constexpr int TILE            = 16;   // edges per async tile
constexpr int WAVES_PER_BLOCK = 4;    // 128 threads/block -> 64KB LDS/block
constexpr int EDGES_PER_WAVE  = 512;  // contiguous chunk per wave

// ---------------------------------------------------------------------------
// Kernel 1: zero the output accumulator (harness poisons d_out, and we
// accumulate sums into it with atomics, so it must start at 0 every launch).
// ---------------------------------------------------------------------------
__global__ void zero_out_kernel(float* __restrict__ out, int n) {
  int i = blockIdx.x * blockDim.x + threadIdx.x;
  int stride = gridDim.x * blockDim.x;
  for (; i < n; i += stride) out[i] = 0.0f;
}

// ---------------------------------------------------------------------------
// Kernel 2: segment-sum of gathered feature rows.
// Each wave processes EDGES_PER_WAVE contiguous (sorted-by-segment) edges.
// Feature rows are gathered into a per-wave LDS double buffer with
// global_load_async_to_lds_b128 (ASYNCcnt-tracked), one instruction per edge:
// lane l moves bytes [16l,16l+16) of the 512B row -> fully coalesced.
// Runs of equal segment id accumulate in registers; boundaries flush with
// global float atomics (segments may span wave boundaries).
// ---------------------------------------------------------------------------
__global__ __launch_bounds__(WAVES_PER_BLOCK * 32) void seg_aggregate_kernel(
    const float* __restrict__ feat, const int* __restrict__ nbr,
    const int* __restrict__ seg, float* __restrict__ out, int n_edges) {
  __shared__ float4 smem[WAVES_PER_BLOCK][2][TILE][32];  // 64 KB

  const int wid  = threadIdx.x >> 5;
  const int lane = threadIdx.x & 31;
  const int wave = blockIdx.x * WAVES_PER_BLOCK + wid;
  const int base = wave * EDGES_PER_WAVE;
  if (base >= n_edges) return;  // wave-uniform exit

  const int count  = min(EDGES_PER_WAVE, n_edges - base);
  const int ntiles = (count + TILE - 1) / TILE;

  const uint64_t feat_base = (uint64_t)(uintptr_t)feat;
  // Generic LDS pointer: low 32 bits = workgroup-relative LDS byte offset,
  // which is exactly what the async instruction's LDS-address VGPR expects.
  const uint32_t lds_base = (uint32_t)(uintptr_t)(&smem[wid][0][0][0]);
  const uint32_t lane_off = (uint32_t)(lane * 16);

  auto issue_tile = [&](int buf, int t) {
    const int tb = base + t * TILE;
#pragma unroll
    for (int e = 0; e < TILE; ++e) {
      int eg = tb + e;
      if (eg > base + count - 1) eg = base + count - 1;  // clamp tail
      const uint32_t row  = (uint32_t)nbr[eg];
      const uint32_t voff = row * (D_FEAT * 4) + lane_off;            // < 2^25
      const uint32_t la   = lds_base +
          (uint32_t)((buf * TILE + e) * (32 * 16)) + lane_off;
      // GVS mode: saddr = features base, vaddr = 32-bit byte offset.
      asm volatile("global_load_async_to_lds_b128 %0, %1, %2"
                   :: "v"(la), "v"(voff), "s"(feat_base)
                   : "memory");
    }
  };

  int    cur = -1;
  float4 acc = make_float4(0.f, 0.f, 0.f, 0.f);

  auto flush = [&]() {
    if (cur >= 0) {
      float* p = out + (size_t)cur * D_FEAT + lane * 4;
      atomicAdd(p + 0, acc.x);
      atomicAdd(p + 1, acc.y);
      atomicAdd(p + 2, acc.z);
      atomicAdd(p + 3, acc.w);
    }
    acc = make_float4(0.f, 0.f, 0.f, 0.f);
  };

  auto process_tile = [&](int buf, int t) {
    const int tb = base + t * TILE;
    const int n  = min(TILE, base + count - tb);
    for (int e = 0; e < n; ++e) {
      const int s = seg[tb + e];         // wave-uniform value (sorted ids)
      if (s != cur) { flush(); cur = s; }
      const float4 v = smem[wid][buf][e][lane];
      acc.x += v.x; acc.y += v.y; acc.z += v.z; acc.w += v.w;
    }
  };

  // Double-buffered software pipeline over ASYNCcnt.
  issue_tile(0, 0);
  for (int t = 0; t < ntiles; ++t) {
    if (t + 1 < ntiles) {
      issue_tile((t + 1) & 1, t + 1);
      // 16 newly issued ops outstanding allowed -> tile t (issued earlier,
      // async loads complete in order) is fully resident in LDS.
      asm volatile("s_wait_asynccnt 16" ::: "memory");
    } else {
      asm volatile("s_wait_asynccnt 0" ::: "memory");
    }
    process_tile(t & 1, t);
  }
  flush();
}

// ---------------------------------------------------------------------------
// Kernel 3: per-node count via binary search on the sorted segment ids
// (no atomics needed), then scale the summed row in place.
// One wave per node; lane owns 4 columns (float4).
// ---------------------------------------------------------------------------
__global__ __launch_bounds__(256) void finalize_kernel(
    const int* __restrict__ seg, float* __restrict__ out,
    int n_nodes, int n_edges) {
  const int wid  = threadIdx.x >> 5;
  const int lane = threadIdx.x & 31;
  const int node = blockIdx.x * (blockDim.x >> 5) + wid;
  if (node >= n_nodes) return;

  // lower_bound(node)
  int lo = 0, hi = n_edges;
  while (lo < hi) {
    int mid = (lo + hi) >> 1;
    if (seg[mid] < node) lo = mid + 1; else hi = mid;
  }
  const int start = lo;
  // upper_bound(node)
  hi = n_edges;
  while (lo < hi) {
    int mid = (lo + hi) >> 1;
    if (seg[mid] <= node) lo = mid + 1; else hi = mid;
  }
  const int cnt = lo - start;
  const float inv = 1.0f / (float)max(cnt, 1);

  float4* row = (float4*)(out + (size_t)node * D_FEAT);
  float4 v = row[lane];
  v.x *= inv; v.y *= inv; v.z *= inv; v.w *= inv;
  row[lane] = v;
}

// ---------------------------------------------------------------------------
extern "C" void kernel_launch(void* const* d_in, const int* in_sizes, int n_in,
                              void* d_out, int out_size, void* d_ws,
                              size_t ws_size, hipStream_t stream) {
  const float* feat = (const float*)d_in[0];
  const int*   nbr  = (const int*)d_in[1];
  const int*   seg  = (const int*)d_in[2];
  float*       out  = (float*)d_out;

  const int n_edges = in_sizes[1];
  const int n_out   = out_size;            // N * 128
  const int n_nodes = n_out / D_FEAT;

  zero_out_kernel<<<1024, 256, 0, stream>>>(out, n_out);

  const int waves  = (n_edges + EDGES_PER_WAVE - 1) / EDGES_PER_WAVE;
  const int blocks = (waves + WAVES_PER_BLOCK - 1) / WAVES_PER_BLOCK;
  seg_aggregate_kernel<<<blocks, WAVES_PER_BLOCK * 32, 0, stream>>>(
      feat, nbr, seg, out, n_edges);

  const int nodes_per_block = 256 / 32;
  finalize_kernel<<<(n_nodes + nodes_per_block - 1) / nodes_per_block, 256, 0,
                    stream>>>(seg, out, n_nodes, n_edges);
}